// ConvolutionalLayer_59219009077551
// MI455X (gfx1250) — compile-verified
//
#include <hip/hip_runtime.h>
#include <hip/hip_bf16.h>

#define F_IN  128
#define F_OUT 128

typedef float v2f __attribute__((ext_vector_type(2)));
typedef float v8f __attribute__((ext_vector_type(8)));

// ---------------------------------------------------------------------------
// Kernel 1: pre_sup = x @ W  via V_WMMA_F32_16X16X4_F32 (fp32-exact)
// Block: 256 threads (8 waves). Block handles a 16-row tile of x (staged in
// LDS once, 8 KB). Wave w computes the 16x16 output tile at columns w*16.
// K-loop: 128 / 4 = 32 WMMA issues per wave.
// ---------------------------------------------------------------------------
__launch_bounds__(256)
__global__ void gcn_gemm_wmma(const float* __restrict__ x,
                              const float* __restrict__ W,
                              float* __restrict__ pre) {
    __shared__ float lds_x[16 * F_IN];           // 16x128 fp32 tile = 8 KB

    const int m0  = blockIdx.x * 16;
    const int tid = threadIdx.x;

    // Cooperative coalesced load of the 16x128 A tile: 512 float4, 2 per thread.
    {
        const float4* src = reinterpret_cast<const float4*>(x + (size_t)m0 * F_IN);
        float4*       dst = reinterpret_cast<float4*>(lds_x);
        dst[tid]       = src[tid];
        dst[tid + 256] = src[tid + 256];
    }
    __syncthreads();

    const int lane  = tid & 31;
    const int wave  = tid >> 5;      // 0..7 -> N tile
    const int n0    = wave * 16;
    const int l16   = lane & 15;
    const int lhalf = lane >> 4;     // 0: K pair {0,1}; 1: K pair {2,3}

    v8f c = {};
    #pragma unroll
    for (int k0 = 0; k0 < F_IN; k0 += 4) {
        const int ka = k0 + lhalf * 2;
        // A fragment (16x4 fp32): lane row = l16, two consecutive K values.
        v2f a;
        a.x = lds_x[l16 * F_IN + ka];
        a.y = lds_x[l16 * F_IN + ka + 1];
        // B fragment (4x16 fp32): lane col = n0+l16, same two K rows of W.
        v2f b;
        b.x = W[(size_t)ka       * F_OUT + n0 + l16];
        b.y = W[(size_t)(ka + 1) * F_OUT + n0 + l16];
        // 8 args: (neg_a, A, neg_b, B, c_mod, C, reuse_a, reuse_b)
        c = __builtin_amdgcn_wmma_f32_16x16x4_f32(
                false, a, false, b, (short)0, c, false, false);
    }

    // C/D layout: VGPR r -> row (lhalf*8 + r), lanes l16 -> col n0+l16.
    #pragma unroll
    for (int r = 0; r < 8; ++r) {
        const int m = m0 + lhalf * 8 + r;
        pre[(size_t)m * F_OUT + n0 + l16] = c[r];
    }
}

// ---------------------------------------------------------------------------
// Kernel 2: out[r] = relu( sum_{e in row r} val[e] * pre[col[e]] + bias )
// adj_rows is sorted -> per-row edge range found by binary search (COO->CSR
// on the fly). One block per row, 128 threads = one output feature each.
// Gathered pre[col] loads are fully coalesced (512 B/edge); no atomics;
// deterministic accumulation order.
// ---------------------------------------------------------------------------
__launch_bounds__(128)
__global__ void gcn_spmm_bias_relu(const int*   __restrict__ rows,
                                   const int*   __restrict__ cols,
                                   const float* __restrict__ vals,
                                   const float* __restrict__ pre,
                                   const float* __restrict__ bias,
                                   float*       __restrict__ out,
                                   int n_edges) {
    const int r   = blockIdx.x;
    const int tid = threadIdx.x;

    __shared__ int s_bounds[2];
    if (tid < 2) {
        const int target = r + tid;          // lower_bound(r), lower_bound(r+1)
        int lo = 0, hi = n_edges;
        while (lo < hi) {
            const int mid = (lo + hi) >> 1;
            if (rows[mid] < target) lo = mid + 1; else hi = mid;
        }
        s_bounds[tid] = lo;
    }
    __syncthreads();

    const int e0 = s_bounds[0];
    const int e1 = s_bounds[1];

    float acc = 0.0f;
    int e = e0;
    // 2-deep manual pipeline on the edge stream for load/FMA overlap.
    for (; e + 1 < e1; e += 2) {
        const int   c0 = cols[e];
        const int   c1 = cols[e + 1];
        const float v0 = vals[e];
        const float v1 = vals[e + 1];
        const float p0 = pre[(size_t)c0 * F_OUT + tid];
        const float p1 = pre[(size_t)c1 * F_OUT + tid];
        acc = fmaf(v0, p0, acc);
        acc = fmaf(v1, p1, acc);
    }
    if (e < e1) {
        acc = fmaf(vals[e], pre[(size_t)cols[e] * F_OUT + tid], acc);
    }

    const float o = acc + bias[tid];
    out[(size_t)r * F_OUT + tid] = o > 0.0f ? o : 0.0f;
}

// ---------------------------------------------------------------------------
// Launch
// ---------------------------------------------------------------------------
extern "C" void kernel_launch(void* const* d_in, const int* in_sizes, int n_in,
                              void* d_out, int out_size, void* d_ws, size_t ws_size,
                              hipStream_t stream) {
    const float* x    = (const float*)d_in[0];   // [N, 128]
    const int*   rows = (const int*)  d_in[1];   // [E] sorted
    const int*   cols = (const int*)  d_in[2];   // [E]
    const float* vals = (const float*)d_in[3];   // [E]
    const float* W    = (const float*)d_in[4];   // [128, 128]
    const float* bias = (const float*)d_in[5];   // [128]

    const int n_nodes = in_sizes[0] / F_IN;      // 100000
    const int n_edges = in_sizes[1];             // 3200000

    float* pre = (float*)d_ws;                   // [N, 128] scratch = 51.2 MB

    // Dense transform: one block per 16-row tile (100000 % 16 == 0).
    gcn_gemm_wmma<<<n_nodes / 16, 256, 0, stream>>>(x, W, pre);

    // Sparse aggregation + bias + ReLU: one block per output row.
    gcn_spmm_bias_relu<<<n_nodes, 128, 0, stream>>>(rows, cols, vals, pre,
                                                    bias, (float*)d_out, n_edges);
}